// GCN_17162689314849
// MI455X (gfx1250) — compile-verified
//
#include <hip/hip_runtime.h>

#define FEAT 128

typedef __attribute__((ext_vector_type(2))) float v2f;
typedef __attribute__((ext_vector_type(8))) float v8f;

// ---------------- zero fill ----------------
__global__ __launch_bounds__(256) void zero_f4(float4* __restrict__ p, int n4) {
  int i = blockIdx.x * blockDim.x + threadIdx.x;
  if (i < n4) p[i] = make_float4(0.f, 0.f, 0.f, 0.f);
}

// ---------------- out[n, 0:64] = bias broadcast ----------------
// 16 float4 chunks per 64-col row; bias is L0-resident.
__global__ __launch_bounds__(256) void init_bias64(
    float4* __restrict__ out, const float4* __restrict__ b4, int total4) {
  int i = blockIdx.x * blockDim.x + threadIdx.x;
  if (i < total4) out[i] = b4[i & 15];
}

// ------------- edge scatter-add, 128-dim messages -------------
// One wave32 per edge: lane L moves 4 contiguous floats (512B coalesced row
// read), then 4 no-return global_atomic_add_f32 (STOREcnt path). Feature
// table (51.2 MB) and accumulator (51.2 MB) are both L2-resident (192 MB L2).
__global__ __launch_bounds__(256) void scatter_add_128(
    const float* __restrict__ feat, const int* __restrict__ src,
    const int* __restrict__ dst, float* __restrict__ agg, int E) {
  int wave = blockIdx.x * (blockDim.x >> 5) + (threadIdx.x >> 5);
  int lane = threadIdx.x & 31;
  if (wave >= E) return;
  int s = src[wave];
  int d = dst[wave];
  const float4 v = *reinterpret_cast<const float4*>(feat + (size_t)s * FEAT + lane * 4);
  float* p = agg + (size_t)d * FEAT + lane * 4;
  atomicAdd(p + 0, v.x);
  atomicAdd(p + 1, v.y);
  atomicAdd(p + 2, v.z);
  atomicAdd(p + 3, v.w);
}

// ------------- edge scatter-add, 64-dim messages (projected) -------------
// segsum(h) @ W2 == segsum(h @ W2): scattering the projected 64-dim rows
// halves the edge-phase bytes and atomic count vs the 128-dim form.
__global__ __launch_bounds__(256) void scatter_add_64(
    const float* __restrict__ feat, const int* __restrict__ src,
    const int* __restrict__ dst, float* __restrict__ agg, int E) {
  int wave = blockIdx.x * (blockDim.x >> 5) + (threadIdx.x >> 5);
  int lane = threadIdx.x & 31;
  if (wave >= E) return;
  int s = src[wave];
  int d = dst[wave];
  const float2 v = *reinterpret_cast<const float2*>(feat + (size_t)s * 64 + lane * 2);
  float* p = agg + (size_t)d * 64 + lane * 2;
  atomicAdd(p + 0, v.x);
  atomicAdd(p + 1, v.y);
}

// ------------- fp32 WMMA GEMM (+optional bias, ReLU) -------------
// out[nrows, NCOL] = A[nrows, 128] @ W[128, NCOL] (+ bias)
// One wave per 16x16 output tile; K loop of V_WMMA_F32_16X16X4_F32.
//
// fp32 A 16x4 fragment (ISA 7.12.2): lane L -> m = L%16, half = L/16;
//   vgpr0 = A[m][k + 2*half], vgpr1 = A[m][k + 2*half + 1]
// fp32 B 4x16 fragment (transpose of A pattern): lane L -> n = L%16;
//   vgpr0 = B[k + 2*half][n], vgpr1 = B[k + 2*half + 1][n]
// fp32 C/D 16x16: lane L -> n = L%16; vgpr i = C[i + 8*half][n]
template <int NCOL, bool RELU, bool BIAS>
__global__ __launch_bounds__(256) void gemm_wmma(
    const float* __restrict__ A, const float* __restrict__ W,
    const float* __restrict__ bias, float* __restrict__ out, int rowTiles) {
  constexpr int COLT = NCOL / 16;
  int tile = blockIdx.x * (blockDim.x >> 5) + (threadIdx.x >> 5);
  if (tile >= rowTiles * COLT) return;   // wave-uniform: EXEC stays all-ones
  int rt = tile / COLT;
  int ct = tile % COLT;
  int lane = threadIdx.x & 31;
  int m = lane & 15;
  int half = lane >> 4;

  const float* arow = A + (size_t)(rt * 16 + m) * FEAT;
  int col = ct * 16 + m;

  v8f acc = {};
#pragma unroll 4
  for (int k = 0; k < FEAT; k += 4) {
    v2f a = *reinterpret_cast<const v2f*>(arow + k + 2 * half);
    v2f b;
    b.x = W[(size_t)(k + 2 * half) * NCOL + col];
    b.y = W[(size_t)(k + 2 * half + 1) * NCOL + col];
    // 8 args: (neg_a, A, neg_b, B, c_mod, C, reuse_a, reuse_b)
    acc = __builtin_amdgcn_wmma_f32_16x16x4_f32(
        false, a, false, b, (short)0, acc, false, false);
  }

  float bv = BIAS ? bias[col] : 0.f;
  float* orow = out + (size_t)(rt * 16 + 8 * half) * NCOL + col;
#pragma unroll
  for (int i = 0; i < 8; ++i) {
    float v = acc[i];
    if (BIAS) v += bv;
    if (RELU) v = fmaxf(v, 0.f);
    orow[(size_t)i * NCOL] = v;
  }
}

extern "C" void kernel_launch(void* const* d_in, const int* in_sizes, int n_in,
                              void* d_out, int out_size, void* d_ws, size_t ws_size,
                              hipStream_t stream) {
  const float* x   = (const float*)d_in[0];
  const int*   src = (const int*)d_in[1];
  const int*   dst = (const int*)d_in[2];
  const float* W1  = (const float*)d_in[3];
  const float* b1  = (const float*)d_in[4];
  const float* W2  = (const float*)d_in[5];
  const float* b2  = (const float*)d_in[6];
  float* out = (float*)d_out;

  const int N = in_sizes[0] / FEAT;   // 100000
  const int E = in_sizes[1];          // 1600000

  float* agg = (float*)d_ws;                  // N*128 floats (layer-1 accum)
  float* h   = agg + (size_t)N * FEAT;        // N*128 floats (layer-1 output)
  float* hp  = agg;                           // N*64 floats (reuse: agg dead after GEMM1)

  const dim3 blk(256);
  const int n4 = N * FEAT / 4;
  const int zeroBlocks = (n4 + 255) / 256;
  const int edgeBlocks = (E + 7) / 8;         // 8 waves / block, 1 edge / wave
  const int rowTiles = N / 16;                // 6250 (exact)

  // ---- layer 1: agg = segsum(x);  h = relu(agg @ W1 + b1) ----
  zero_f4<<<zeroBlocks, blk, 0, stream>>>((float4*)agg, n4);
  scatter_add_128<<<edgeBlocks, blk, 0, stream>>>(x, src, dst, agg, E);
  {
    int tiles = rowTiles * (FEAT / 16);
    gemm_wmma<FEAT, true, true>
        <<<(tiles + 7) / 8, blk, 0, stream>>>(agg, W1, b1, h, rowTiles);
  }

  // ---- layer 2 (project-first): hp = h @ W2;  out = b2 + segsum(hp) ----
  {
    int tiles = rowTiles * (64 / 16);
    gemm_wmma<64, false, false>
        <<<(tiles + 7) / 8, blk, 0, stream>>>(h, W2, nullptr, hp, rowTiles);
  }
  {
    int total4 = N * 64 / 4;
    init_bias64<<<(total4 + 255) / 256, blk, 0, stream>>>(
        (float4*)out, (const float4*)b2, total4);
  }
  scatter_add_64<<<edgeBlocks, blk, 0, stream>>>(hp, src, dst, out, E);
}